// HarmonicModel_38774964748560
// MI455X (gfx1250) — compile-verified
//
#include <hip/hip_runtime.h>
#include <hip/hip_bf16.h>
#include <math.h>

// Problem constants
#define NFRM   128      // frames
#define NVOICE 8
#define NPROF  16
#define NHARM  64
#define NPART  65       // 1 fundamental + 64 harmonics
#define NSAMP  32768
#define NBATCH 8
#define NVP    (NVOICE * NPART)   // 520 (voice,partial) pairs per batch

static __device__ constexpr float kMinFreq = 40.0f / 11025.0f;   // 40/NYQUIST
static __device__ constexpr float kPi      = 3.14159265358979323846f;

typedef __attribute__((ext_vector_type(2))) float v2f;
typedef __attribute__((ext_vector_type(8))) float v8f;

// ---------------------------------------------------------------------------
// Kernel 1: per (batch,voice) block.
//   Phase A : freq / f0_amp per frame, softmax over 16 profile logits -> LDS.
//   Phase B : mix(128x16) @ profiles(16x64) via v_wmma_f32_16x16x4_f32;
//             harmonic freq rows; write FREQ / AMP rows (65 x 128) to ws.
//   Phase C : frame-level prefix C(k) of the piecewise-linear cumsum.
// ---------------------------------------------------------------------------
__global__ __launch_bounds__(256) void harmonic_prep_kernel(
    const float* __restrict__ f0,       // [64][2][128]
    const float* __restrict__ harm,     // [64][16][128]
    const float* __restrict__ prof,     // [16][64]
    float* __restrict__ FREQ,           // [64][65][128]
    float* __restrict__ AMP,            // [64][65][128]
    float* __restrict__ CPS)            // [64][65][128]
{
    __shared__ float smix[NFRM][NPROF];   // softmax mixture, 8 KB
    __shared__ float sprof[NPROF][NHARM]; // profiles, 4 KB
    __shared__ float sff[NPART][NFRM];    // freq rows, 32.5 KB
    __shared__ float samp[NFRM];          // f0 amplitude

    const int bv  = blockIdx.x;           // 0..63  (= b*8 + v)
    const int tid = threadIdx.x;

    // ---------------- Phase A ----------------
    if (tid < NFRM) {
        const int f = tid;
        const float* fptr = f0 + (size_t)bv * 2 * NFRM;
        const float re = fptr[f];
        const float im = fptr[NFRM + f];
        samp[f] = re * re + im * im;
        const float ang = atan2f(im, re) * (1.0f / kPi);
        sff[0][f] = kMinFreq + ang * ang * kMinFreq;

        // softmax over the 16 profile logits for this frame
        const float* hptr = harm + (size_t)bv * NPROF * NFRM + f;
        float h[NPROF];
        float m = -INFINITY;
        #pragma unroll
        for (int p = 0; p < NPROF; ++p) { h[p] = hptr[(size_t)p * NFRM]; m = fmaxf(m, h[p]); }
        float ssum = 0.0f;
        #pragma unroll
        for (int p = 0; p < NPROF; ++p) { h[p] = __expf(h[p] - m); ssum += h[p]; }
        const float inv = 1.0f / ssum;
        #pragma unroll
        for (int p = 0; p < NPROF; ++p) smix[f][p] = h[p] * inv;
    } else {
        // other 128 threads stage the 16x64 profile matrix
        const int t = tid - 128;
        #pragma unroll
        for (int r = 0; r < 8; ++r) {
            const int idx = t + r * 128;
            ((float*)sprof)[idx] = prof[idx];
        }
    }
    __syncthreads();

    float* FREQb = FREQ + (size_t)bv * NPART * NFRM;
    float* AMPb  = AMP  + (size_t)bv * NPART * NFRM;

    // ---------------- Phase B: harmonic freq rows ----------------
    for (int idx = tid; idx < NHARM * NFRM; idx += 256) {
        const int h = idx >> 7;
        const int f = idx & 127;
        const float ratio = (float)((h + 2) * (h + 2));
        const float hf = fminf(sff[0][f] * ratio, 1.0f);
        sff[1 + h][f] = hf;
        FREQb[(size_t)(1 + h) * NFRM + f] = hf;
    }

    // ---------------- Phase B: WMMA GEMM  mix(128x16) @ prof(16x64) --------
    // Each wave owns a 16-frame M-tile; 4 N-tiles of 16; K=16 as 4 chunks of 4.
    const int wave  = tid >> 5;
    const int lane  = tid & 31;
    const int lo    = lane & 15;
    const int hi    = lane >> 4;          // 0 or 1
    const int fbase = wave << 4;

    v8f acc[4];
    #pragma unroll
    for (int n = 0; n < 4; ++n) acc[n] = (v8f){0.f,0.f,0.f,0.f,0.f,0.f,0.f,0.f};

    #pragma unroll
    for (int kc = 0; kc < NPROF; kc += 4) {
        // A 16x4 f32 fragment: VGPR0 holds K=kc(+2*hi), VGPR1 holds K=kc+1(+2*hi)
        v2f a;
        a.x = smix[fbase + lo][kc + 2 * hi + 0];
        a.y = smix[fbase + lo][kc + 2 * hi + 1];
        #pragma unroll
        for (int n = 0; n < 4; ++n) {
            // B 4x16 f32 fragment: rows kc..kc+3 striped across lane halves
            v2f bf;
            bf.x = sprof[kc + 0 + hi][(n << 4) + lo];
            bf.y = sprof[kc + 2 + hi][(n << 4) + lo];
            acc[n] = __builtin_amdgcn_wmma_f32_16x16x4_f32(
                false, a, false, bf, (short)0, acc[n], false, false);
        }
    }

    // D layout: VGPR r -> (M = r + 8*hi, N = lo).  hamp = clip(.,0,1) * f0_amp
    #pragma unroll
    for (int n = 0; n < 4; ++n) {
        #pragma unroll
        for (int r = 0; r < 8; ++r) {
            const int f = fbase + r + 8 * hi;
            const int h = (n << 4) + lo;
            const float val = fminf(fmaxf(acc[n][r], 0.0f), 1.0f) * samp[f];
            AMPb[(size_t)(1 + h) * NFRM + f] = val;
        }
    }

    // fundamental rows (p = 0)
    if (tid < NFRM) {
        FREQb[tid] = sff[0][tid];
        AMPb[tid]  = samp[tid];
    }
    __syncthreads();

    // ---------------- Phase C: frame-level cumsum bases ----------------
    // C(0) = 128*x0 (head plateau); C(k) = C(k-1) + 128*(x_{k-1} + x_k)
    if (tid < NPART) {
        float* Crow = CPS + (size_t)bv * NPART * NFRM + (size_t)tid * NFRM;
        float prev = sff[tid][0];
        float c = 128.0f * prev;
        Crow[0] = c;
        for (int k = 1; k < NFRM; ++k) {
            const float x = sff[tid][k];
            c += 128.0f * (prev + x);
            Crow[k] = c;
            prev = x;
        }
    }
}

// ---------------------------------------------------------------------------
// Kernel 2: synthesis.  Grid (129, 8): blockIdx.x = segment (0 = head plateau,
// 1..127 = linear ramp k-1, 128 = tail plateau), blockIdx.y = batch.
// Each block stages 520 (voice,partial) records {x, dx, a, da, C} in LDS,
// then each thread evaluates its sample:
//   S(j) = C + (j+1)*x + dx*(j+1)^2/512 ;  amp(j) = a + da*(j+0.5)/256
//   out += amp * sin(pi*S)   via v_sin_f32 on frac(S/2) revolutions.
// ---------------------------------------------------------------------------
__global__ __launch_bounds__(256) void harmonic_synth_kernel(
    const float* __restrict__ FREQ,
    const float* __restrict__ AMP,
    const float* __restrict__ CPS,
    float* __restrict__ out)            // [8][32768]
{
    __shared__ float4 rec[NVP];   // {x, dx, a, da}
    __shared__ float  cb[NVP];    // cumsum base C

    const int s   = blockIdx.x;   // 0..128
    const int b   = blockIdx.y;   // 0..7
    const int tid = threadIdx.x;

    for (int vp = tid; vp < NVP; vp += 256) {
        const int v = vp / NPART;
        const int p = vp - v * NPART;
        const size_t base = ((size_t)(b * NVOICE + v) * NPART + p) * NFRM;
        float x, dx, a, da, C;
        if (s == 0) {                       // head plateau: value = x[0]
            x = FREQ[base]; dx = 0.0f;
            a = AMP[base];  da = 0.0f;
            C = 0.0f;
        } else if (s < 128) {               // ramp k = s-1
            const int k = s - 1;
            const float xk  = FREQ[base + k];
            const float xk1 = FREQ[base + k + 1];
            const float ak  = AMP[base + k];
            const float ak1 = AMP[base + k + 1];
            x = xk;  dx = xk1 - xk;
            a = ak;  da = ak1 - ak;
            C = CPS[base + k];
        } else {                            // tail plateau: value = x[127]
            x = FREQ[base + 127]; dx = 0.0f;
            a = AMP[base + 127];  da = 0.0f;
            C = CPS[base + 127];
        }
        rec[vp] = make_float4(x, dx, a, da);
        cb[vp]  = C;
    }
    __syncthreads();

    const int   j  = tid;
    const float jf = (float)(j + 1);
    const float w  = ((float)j + 0.5f) * (1.0f / 256.0f);
    const float q  = jf * jf * (1.0f / 512.0f);

    float sum = 0.0f;
    #pragma unroll 4
    for (int vp = 0; vp < NVP; ++vp) {
        const float4 r = rec[vp];                      // LDS broadcast
        const float  S = cb[vp] + jf * r.x + q * r.y;  // inclusive cumsum
        const float  amp = r.z + w * r.w;
        float u = 0.5f * S;                            // revolutions: sin(pi*S)=sin(2pi*u)
        u = u - floorf(u);
        sum += amp * __builtin_amdgcn_sinf(u);         // v_sin_f32
    }

    const bool valid = (s >= 1 && s <= 127) || (tid < 128);
    const int  i = (s == 0) ? tid : ((s < 128) ? (256 * s - 128 + tid) : (32640 + tid));
    if (valid) out[(size_t)b * NSAMP + i] = sum;
}

// ---------------------------------------------------------------------------
// Workspace layout (floats): FREQ[64*65*128] | AMP[64*65*128] | CPS[64*65*128]
// = 6,389,760 bytes total.
// ---------------------------------------------------------------------------
extern "C" void kernel_launch(void* const* d_in, const int* in_sizes, int n_in,
                              void* d_out, int out_size, void* d_ws, size_t ws_size,
                              hipStream_t stream) {
    const float* f0   = (const float*)d_in[0];   // (8,8,2,128)   f32
    const float* harm = (const float*)d_in[1];   // (8,8,16,128)  f32
    const float* prof = (const float*)d_in[2];   // (16,64)       f32

    const size_t rowset = (size_t)NBATCH * NVOICE * NPART * NFRM;  // 532480
    float* FREQ = (float*)d_ws;
    float* AMPw = FREQ + rowset;
    float* CPS  = AMPw + rowset;

    harmonic_prep_kernel<<<NBATCH * NVOICE, 256, 0, stream>>>(
        f0, harm, prof, FREQ, AMPw, CPS);

    harmonic_synth_kernel<<<dim3(129, NBATCH), 256, 0, stream>>>(
        FREQ, AMPw, CPS, (float*)d_out);
}